// Decoder2_67525475828263
// MI455X (gfx1250) — compile-verified
//
#include <hip/hip_runtime.h>

typedef __attribute__((ext_vector_type(2))) float v2f;
typedef __attribute__((ext_vector_type(8))) float v8f;

// explicit global-memory pointers so weight traffic lowers to global_load_*
// (FLAT would consume DScnt too and tie up the LDS path; ISA §10.1.2)
typedef __attribute__((address_space(1))) const float* gcf;
typedef __attribute__((address_space(1))) float*       gf;

#define H   100
#define H2  50
#define CP  25
#define MP  130
#define NK  24
#define NM  7
#define NTHREADS 256
#define NWAVES   8

// ---- output layout (flat, in tuple return order) ----
#define OUT_CHORD   0              // 50*25 = 1250
#define OUT_MEL     1250           // 400*130 = 52000
#define OUT_KEY     53250          // 24
#define OUT_MODE    53274          // 7
#define OUT_BPM     53281
#define OUT_ENERGY  53282
#define OUT_VALENCE 53283

struct Params {
    const float* p[45];
    float* out;
};

enum {
    I_Z = 0,
    I_CL_WIH, I_CL_WHH, I_CL_BIH, I_CL_BHH,
    I_CE_W, I_CE_B,
    I_CP_W1, I_CP_B1, I_CP_W2, I_CP_B2,
    I_CD_W, I_CD_B,
    I_ME_W, I_ME_B,
    I_ML_WIH, I_ML_WHH, I_ML_BIH, I_ML_BHH,
    I_MP_W1, I_MP_B1, I_MP_W2, I_MP_B2,
    I_MD_W, I_MD_B,
    I_KEY_W1, I_KEY_B1, I_KEY_W2, I_KEY_B2,
    I_MODE_W1, I_MODE_B1, I_MODE_W2, I_MODE_B2,
    I_TP_W1, I_TP_B1, I_TP_W2, I_TP_B2,
    I_VL_W1, I_VL_B1, I_VL_W2, I_VL_B2,
    I_EN_W1, I_EN_B1, I_EN_W2, I_EN_B2
};

__device__ __forceinline__ float sigmoidf(float x) { return 1.0f / (1.0f + __expf(-x)); }
__device__ __forceinline__ float clip01(float x)   { return fminf(fmaxf(x, 0.0f), 1.0f); }

__device__ __forceinline__ v8f wmma_f32(v2f a, v2f b, v8f c) {
    return __builtin_amdgcn_wmma_f32_16x16x4_f32(false, a, false, b, (short)0, c, false, false);
}

// One K-pass of y_tile += W[16 rows x K] * x, via V_WMMA_F32_16X16X4_F32.
// A layout (ISA): lane l<16 holds row m=l, K cols {k0, k0+1}; lane l+16 holds
// row m=l-16, K cols {k0+2, k0+3}. B = x broadcast across all 16 N-columns.
// GM: last-tile row masking — row index is CLAMPED by the caller so all loads
// are in-bounds, masking is a branch-free select (no EXEC divergence).
// Ragged K tail: W index clamped + select-masked; x reads rely on the matching
// A element being 0 (x buffers are finite/zero-padded to a multiple of 4).
// Two accumulators halve the serial WMMA->WMMA dependency chain.
template<int K, bool GM>
static __device__ __forceinline__ void accum_pass(const float* __restrict__ W, int mc, bool mok,
                                                  const float* __restrict__ x, int kc,
                                                  v8f& acc0, v8f& acc1)
{
    const gcf Wr = (gcf)(W + (size_t)mc * (size_t)K);   // global_load path
    constexpr int KF = K & ~3;
    #pragma unroll 4
    for (int k0 = 0; k0 < KF; k0 += 4) {
        float w0 = Wr[k0 + kc];          // one global_load_b64 per k-step
        float w1 = Wr[k0 + kc + 1];
        if constexpr (GM) { w0 = mok ? w0 : 0.0f; w1 = mok ? w1 : 0.0f; }
        v2f a, b;
        a.x = w0; a.y = w1;
        b.x = x[k0 + kc];                // ds_load path (broadcast across 16 lanes)
        b.y = x[k0 + kc + 1];
        if ((k0 >> 2) & 1) acc1 = wmma_f32(a, b, acc1);
        else               acc0 = wmma_f32(a, b, acc0);
    }
    if constexpr (KF != K) {             // ragged-K tail, branch-free
        const int k   = KF + kc;
        const int kl0 = (k     < K) ? k     : (K - 1);
        const int kl1 = (k + 1 < K) ? k + 1 : (K - 1);
        float w0 = Wr[kl0];
        float w1 = Wr[kl1];
        const bool ok0 = (k     < K) && (!GM || mok);
        const bool ok1 = (k + 1 < K) && (!GM || mok);
        v2f a, b;
        a.x = ok0 ? w0 : 0.0f;
        a.y = ok1 ? w1 : 0.0f;
        b.x = x[k];                      // padded region: finite, masked by a==0
        b.y = x[k + 1];
        acc0 = wmma_f32(a, b, acc0);
    }
}

// y[n] = act( W1[n,:].x1 + (K2? W2[n,:].x2 : 0) + b1[n] + (K2? b2[n] : 0) )
// W row-major [N x K]. Executed cooperatively by all 8 waves; M-tiles are
// round-robined across waves. Caller syncs before/after.
template<int N, int K1, int K2, bool RELU>
static __device__ void matvec(const float* __restrict__ W1, const float* __restrict__ x1,
                              const float* __restrict__ W2, const float* __restrict__ x2,
                              const float* __restrict__ b1, const float* __restrict__ b2,
                              float* __restrict__ y)
{
    const int lane = threadIdx.x & 31;
    const int wave = threadIdx.x >> 5;
    const int mrow = lane & 15;
    const int kc   = (lane >> 4) * 2;
    constexpr int NTILES = (N + 15) / 16;
    constexpr bool NTAIL = (N % 16) != 0;

    for (int t = wave; t < NTILES; t += NWAVES) {
        const int m = t * 16 + mrow;
        v8f a0 = {}, a1 = {};
        if (!NTAIL || (t != NTILES - 1)) {     // full tiles: zero guards at all
            accum_pass<K1, false>(W1, m, true, x1, kc, a0, a1);
            if constexpr (K2 > 0) accum_pass<K2, false>(W2, m, true, x2, kc, a0, a1);
        } else {                                // tail tile: clamped rows + select mask
            const bool mok = (m < N);
            const int  mc  = mok ? m : (N - 1);
            accum_pass<K1, true>(W1, mc, mok, x1, kc, a0, a1);
            if constexpr (K2 > 0) accum_pass<K2, true>(W2, mc, mok, x2, kc, a0, a1);
        }
        // D: lane l<16 holds row=vgpr col=l; lane 16+l holds row=vgpr+8.
        // All columns equal (B broadcast): lane 0 -> rows 0..7, lane 16 -> rows 8..15.
        if ((lane & 15) == 0) {
            const gcf b1g = (gcf)b1;
            const gcf b2g = (gcf)b2;
            const int base = t * 16 + ((lane >> 4) ? 8 : 0);
            #pragma unroll
            for (int r = 0; r < 8; ++r) {
                const int n = base + r;
                if (n < N) {
                    float v = a0[r] + a1[r] + b1g[n];
                    if constexpr (K2 > 0) v += b2g[n];
                    if constexpr (RELU) v = fmaxf(v, 0.0f);
                    y[n] = v;
                }
            }
        }
    }
}

__device__ __forceinline__ void lstm_update(const float* __restrict__ gates,
                                            float* __restrict__ h, float* __restrict__ c)
{
    const int t = threadIdx.x;
    if (t < H) {
        const float ig = sigmoidf(gates[t]);
        const float fg = sigmoidf(gates[H + t]);
        const float gg = tanhf(gates[2 * H + t]);
        const float og = sigmoidf(gates[3 * H + t]);
        const float cn = fg * c[t] + ig * gg;
        c[t] = cn;
        h[t] = og * tanhf(cn);
    }
}

__global__ __launch_bounds__(NTHREADS)
void decoder2_persistent(Params P)
{
    __shared__ float s_z[H];
    __shared__ float s_hc[H], s_cc[H], s_hm[H], s_cm[H];
    __shared__ float s_chemb[H], s_memb[H];
    __shared__ float s_gates[4 * H];
    __shared__ float s_hid[H];        // MLP hidden (K=100, no pad needed)
    __shared__ float s_pred[MP + 2];  // chord/mel pred; padded for ragged-K reads
    __shared__ float s_tmp[H];        // ce / me intermediate, head scalars
    __shared__ float s_cat[3 * H];    // concat input (200/300, K%4==0)
    __shared__ float s_head[H2 + 2];  // head hidden; padded for K=50 tail

    const int tid = threadIdx.x;
    float* out = P.out;
    gf outg = (gf)P.out;

    // ---- init state (zero pads so ragged-K reads are finite) ----
    if (tid < H) {
        const float zv = ((gcf)P.p[I_Z])[tid];
        s_z[tid] = zv;
        s_hc[tid] = 0.0f; s_cc[tid] = 0.0f;
        s_hm[tid] = 0.0f; s_cm[tid] = 0.0f;
        s_chemb[tid] = zv;            // outer carry starts with ch_emb = z
        s_memb[tid] = 0.0f;
    }
    if (tid < MP + 2) s_pred[tid] = 0.0f;
    if (tid < H2 + 2) s_head[tid] = 0.0f;
    __syncthreads();

    // ---- heads (independent of the recurrence) ----
    matvec<H2, H, 0, true>(P.p[I_KEY_W1], s_z, nullptr, nullptr, P.p[I_KEY_B1], nullptr, s_head);
    __syncthreads();
    matvec<NK, H2, 0, false>(P.p[I_KEY_W2], s_head, nullptr, nullptr, P.p[I_KEY_B2], nullptr, out + OUT_KEY);
    __syncthreads();
    matvec<H2, H, 0, true>(P.p[I_MODE_W1], s_z, nullptr, nullptr, P.p[I_MODE_B1], nullptr, s_head);
    __syncthreads();
    matvec<NM, H2, 0, false>(P.p[I_MODE_W2], s_head, nullptr, nullptr, P.p[I_MODE_B2], nullptr, out + OUT_MODE);
    __syncthreads();
    matvec<H2, H, 0, true>(P.p[I_TP_W1], s_z, nullptr, nullptr, P.p[I_TP_B1], nullptr, s_head);
    __syncthreads();
    matvec<1, H2, 0, false>(P.p[I_TP_W2], s_head, nullptr, nullptr, P.p[I_TP_B2], nullptr, s_tmp);
    __syncthreads();
    if (tid == 0) outg[OUT_BPM] = clip01(s_tmp[0]) * 30.0f + 70.0f;
    __syncthreads();
    matvec<H2, H, 0, true>(P.p[I_VL_W1], s_z, nullptr, nullptr, P.p[I_VL_B1], nullptr, s_head);
    __syncthreads();
    matvec<1, H2, 0, false>(P.p[I_VL_W2], s_head, nullptr, nullptr, P.p[I_VL_B2], nullptr, s_tmp);
    __syncthreads();
    if (tid == 0) outg[OUT_VALENCE] = clip01(s_tmp[0]);
    __syncthreads();
    matvec<H2, H, 0, true>(P.p[I_EN_W1], s_z, nullptr, nullptr, P.p[I_EN_B1], nullptr, s_head);
    __syncthreads();
    matvec<1, H2, 0, false>(P.p[I_EN_W2], s_head, nullptr, nullptr, P.p[I_EN_B2], nullptr, s_tmp);
    __syncthreads();
    if (tid == 0) outg[OUT_ENERGY] = clip01(s_tmp[0]);
    __syncthreads();

    // ---- recurrence: 50 outer x 8 inner steps, one persistent workgroup ----
    for (int i = 0; i < 50; ++i) {
        // chord LSTM gates = ch_emb @ Wih^T + bih + hc @ Whh^T + bhh (fused)
        matvec<4 * H, H, H, false>(P.p[I_CL_WIH], s_chemb, P.p[I_CL_WHH], s_hc,
                                   P.p[I_CL_BIH], P.p[I_CL_BHH], s_gates);
        __syncthreads();
        lstm_update(s_gates, s_hc, s_cc);
        __syncthreads();

        // chord prediction MLP
        matvec<H, H, 0, true>(P.p[I_CP_W1], s_hc, nullptr, nullptr, P.p[I_CP_B1], nullptr, s_hid);
        __syncthreads();
        matvec<CP, H, 0, false>(P.p[I_CP_W2], s_hid, nullptr, nullptr, P.p[I_CP_B2], nullptr, s_pred);
        __syncthreads();
        if (tid < CP) outg[OUT_CHORD + i * CP + tid] = s_pred[tid];

        // ce = chord_pred @ ce_w^T + ce_b   (K=25 ragged tail, select-masked)
        matvec<H, CP, 0, false>(P.p[I_CE_W], s_pred, nullptr, nullptr, P.p[I_CE_B], nullptr, s_tmp);
        __syncthreads();
        // ch_emb = concat(ce, z) @ cd_w^T + cd_b
        if (tid < H) { s_cat[tid] = s_tmp[tid]; s_cat[H + tid] = s_z[tid]; }
        __syncthreads();
        matvec<H, 2 * H, 0, false>(P.p[I_CD_W], s_cat, nullptr, nullptr, P.p[I_CD_B], nullptr, s_chemb);
        __syncthreads();

        if (i == 0) {
            if (tid < H) s_memb[tid] = s_chemb[tid];
            __syncthreads();
        }

        for (int j = 0; j < 8; ++j) {
            // melody LSTM (fused dual matvec)
            matvec<4 * H, H, H, false>(P.p[I_ML_WIH], s_memb, P.p[I_ML_WHH], s_hm,
                                       P.p[I_ML_BIH], P.p[I_ML_BHH], s_gates);
            __syncthreads();
            lstm_update(s_gates, s_hm, s_cm);
            __syncthreads();

            // melody prediction MLP
            matvec<H, H, 0, true>(P.p[I_MP_W1], s_hm, nullptr, nullptr, P.p[I_MP_B1], nullptr, s_hid);
            __syncthreads();
            matvec<MP, H, 0, false>(P.p[I_MP_W2], s_hid, nullptr, nullptr, P.p[I_MP_B2], nullptr, s_pred);
            __syncthreads();
            if (tid < MP) outg[OUT_MEL + (i * 8 + j) * MP + tid] = s_pred[tid];

            // me = mel_pred @ me_w^T + me_b  (K=130 ragged tail)
            matvec<H, MP, 0, false>(P.p[I_ME_W], s_pred, nullptr, nullptr, P.p[I_ME_B], nullptr, s_tmp);
            __syncthreads();
            // me = concat(me, ch_emb, z) @ md_w^T + md_b
            if (tid < H) {
                s_cat[tid]         = s_tmp[tid];
                s_cat[H + tid]     = s_chemb[tid];
                s_cat[2 * H + tid] = s_z[tid];
            }
            __syncthreads();
            matvec<H, 3 * H, 0, false>(P.p[I_MD_W], s_cat, nullptr, nullptr, P.p[I_MD_B], nullptr, s_memb);
            __syncthreads();
        }
    }
}

extern "C" void kernel_launch(void* const* d_in, const int* in_sizes, int n_in,
                              void* d_out, int out_size, void* d_ws, size_t ws_size,
                              hipStream_t stream) {
    (void)in_sizes; (void)d_ws; (void)ws_size; (void)out_size;
    Params P;
    for (int k = 0; k < 45 && k < n_in; ++k) P.p[k] = (const float*)d_in[k];
    P.out = (float*)d_out;
    decoder2_persistent<<<1, NTHREADS, 0, stream>>>(P);
}